// CRF_17145509445684
// MI455X (gfx1250) — compile-verified
//
#include <hip/hip_runtime.h>
#include <stdint.h>

// Problem constants (from reference): B=256, L=2048, T=48
#define Bn 256
#define Ln 2048
#define Tn 48
#define CH 32                         // time-steps per TDM chunk
#define NSTEPS (Ln - 1)               // 2047 DP steps (t = 1 .. L-1)
#define NCH ((NSTEPS + CH - 1) / CH)  // 64 chunks
#define NSEG 64                       // backtrace segments
#define SEGLEN 32                     // steps per segment (64*32 >= 2047)
#define NTHREADS 192                  // 6 waves: 4 reduction groups x 48 tags

typedef uint32_t u32;
typedef __attribute__((ext_vector_type(4))) u32 u32x4;
typedef __attribute__((ext_vector_type(8))) u32 u32x8;

// ---- dynamic LDS layout (bytes) ----
#define OFF_TRANS  0                               // 48*48 f32 = 9216
#define OFF_SBUF   (Tn * Tn * 4)                   // 2*48 f32 ping-pong score
#define OFF_PVAL   (OFF_SBUF + 2 * Tn * 4)         // 4*48 f32 partial max
#define OFF_PIDX   (OFF_PVAL + 4 * Tn * 4)         // 4*48 i32 partial argmax
#define OFF_EIN    (OFF_PIDX + 4 * Tn * 4)         // NSEG i32 boundary tags
#define OFF_PBUF   (OFF_EIN + NSEG * 4)            // 2*CH*48 f32 TDM ring
#define OFF_SEGMAP (OFF_PBUF + 2 * CH * Tn * 4)    // NSEG*48 u8 segment maps
#define OFF_BP     (OFF_SEGMAP + NSEG * Tn)        // 2047*48 u8 backptrs
#define LDS_BYTES  (OFF_BP + NSTEPS * Tn + 64)     // ~125 KB < 320 KB WGP LDS

// Issue a 1-D TDM load of `nelem` f32 from global -> LDS (async, TENSORcnt).
// D# group0/group1 built per CDNA5 ISA ch.8; VADDR2/VADDR3 omitted (<=2D).
__device__ __forceinline__ void tdm_load_f32(u32 lds_byte_off,
                                             const float* gsrc, u32 nelem) {
  uint64_t ga = (uint64_t)(uintptr_t)gsrc;
  u32x4 g0;
  g0.x = 1u;                                    // count=1, user descriptor
  g0.y = lds_byte_off;                          // LDS byte address
  g0.z = (u32)ga;                               // global_addr[31:0]
  g0.w = (u32)((ga >> 32) & 0x01FFFFFFu)        // global_addr[56:32]
         | (2u << 30);                          // type = 2 ("image")
  u32x8 g1;
  g1.s0 = (2u << 16);                           // data_size = 4 bytes
  g1.s1 = (nelem & 0xFFFFu) << 16;              // tensor_dim0[15:0]
  g1.s2 = (nelem >> 16) | (1u << 16);           // tensor_dim0[31:16], dim1=1
  g1.s3 = (nelem & 0xFFFFu) << 16;              // tile_dim0 = nelem (<=1536)
  g1.s4 = 1u;                                   // tile_dim1 = 1, tile_dim2 = 0
  g1.s5 = nelem;                                // tensor_dim0_stride[31:0]
  g1.s6 = 0u;
  g1.s7 = 0u;
  asm volatile("tensor_load_to_lds %0, %1" :: "s"(g0), "s"(g1) : "memory");
}

// First-occurrence argmax combine: `a` carries the lower index; keep a on tie.
__device__ __forceinline__ void comb(float va, int ia, float vb, int ib,
                                     float& vo, int& io) {
  const bool take_b = vb > va;
  vo = take_b ? vb : va;
  io = take_b ? ib : ia;
}

__global__ __launch_bounds__(NTHREADS)
void crf_viterbi_kernel(const float* __restrict__ pot,
                        const int* __restrict__ lengths,
                        const float* __restrict__ trans,
                        int* __restrict__ out_tags,
                        float* __restrict__ out_trans) {
  extern __shared__ char smem[];
  float* trans_s = (float*)(smem + OFF_TRANS);
  float* sbuf    = (float*)(smem + OFF_SBUF);
  float* pval    = (float*)(smem + OFF_PVAL);
  int*   pidx    = (int*)(smem + OFF_PIDX);
  int*   ein     = (int*)(smem + OFF_EIN);
  float* pbuf    = (float*)(smem + OFF_PBUF);
  unsigned char* segmap = (unsigned char*)(smem + OFF_SEGMAP);
  unsigned char* bp     = (unsigned char*)(smem + OFF_BP);
  const u32 lds_pbuf = (u32)(uintptr_t)(smem + OFF_PBUF);

  const int tid = threadIdx.x;
  const int b   = blockIdx.x;
  const int q   = tid / Tn;                 // reduction group 0..3
  const int j   = tid % Tn;                 // tag 0..47
  const int len = lengths[b];
  const float* potb = pot + (size_t)b * Ln * Tn;

  // Stage transitions in LDS (coalesced); block 0 also emits output #2.
  for (int idx = tid; idx < Tn * Tn; idx += NTHREADS) trans_s[idx] = trans[idx];
  if (b == 0)
    for (int idx = tid; idx < Tn * Tn; idx += NTHREADS)
      out_trans[idx] = trans[idx];

  // init_score = potentials[b, 0, :]
  if (tid < Tn) sbuf[tid] = potb[tid];
  __syncthreads();

  // Each thread owns 12 source tags i = q*12 .. q*12+11 for column j:
  const int ibase = q * 12;
  float rt[12];
#pragma unroll
  for (int m = 0; m < 12; ++m) rt[m] = trans_s[(ibase + m) * Tn + j];

  const bool wave0 = (tid < 32);            // TDM issued once (by wave 0)
  if (wave0) tdm_load_f32(lds_pbuf, potb + (size_t)1 * Tn, CH * Tn);

  int cur = 0;
  for (int k = 0; k < NCH; ++k) {
    const int t0 = 1 + k * CH;
    const int rem = NSTEPS - k * CH;
    const int csteps = rem < CH ? rem : CH;

    if (wave0) {
      if (k + 1 < NCH) {                    // prefetch next chunk, then wait
        const int nt0 = 1 + (k + 1) * CH;   // for the current one (in-order)
        const int nrem = NSTEPS - (k + 1) * CH;
        const int ns = nrem < CH ? nrem : CH;
        tdm_load_f32(lds_pbuf + (u32)(((k + 1) & 1) * CH * Tn * 4),
                     potb + (size_t)nt0 * Tn, (u32)(ns * Tn));
        __builtin_amdgcn_s_wait_tensorcnt(1);
      } else {
        __builtin_amdgcn_s_wait_tensorcnt(0);
      }
    }
    __syncthreads();

    const float* pc = pbuf + (k & 1) * CH * Tn;
    for (int s = 0; s < csteps; ++s) {
      const int t = t0 + s;
      {
        // Partial reduce: 12 candidates, depth-4 tree, first-occurrence ties.
        const float* sc = sbuf + cur * Tn + ibase;
        const float4 f0 = *(const float4*)(sc + 0);
        const float4 f1 = *(const float4*)(sc + 4);
        const float4 f2 = *(const float4*)(sc + 8);
        const float c[12] = {
            f0.x + rt[0], f0.y + rt[1], f0.z + rt[2],  f0.w + rt[3],
            f1.x + rt[4], f1.y + rt[5], f1.z + rt[6],  f1.w + rt[7],
            f2.x + rt[8], f2.y + rt[9], f2.z + rt[10], f2.w + rt[11]};
        float v6[6]; int i6[6];
#pragma unroll
        for (int m = 0; m < 6; ++m)
          comb(c[2 * m], 2 * m, c[2 * m + 1], 2 * m + 1, v6[m], i6[m]);
        float v3[3]; int i3[3];
#pragma unroll
        for (int m = 0; m < 3; ++m)
          comb(v6[2 * m], i6[2 * m], v6[2 * m + 1], i6[2 * m + 1], v3[m], i3[m]);
        float va; int ia;
        comb(v3[0], i3[0], v3[1], i3[1], va, ia);
        float vb; int ib;
        comb(va, ia, v3[2], i3[2], vb, ib);
        pval[q * Tn + j] = vb;
        pidx[q * Tn + j] = ibase + ib;
      }
      __syncthreads();
      if (tid < Tn) {
        // Combine 4 partials in ascending-q order (keeps lowest index on tie).
        float vA; int iA;
        comb(pval[tid], pidx[tid], pval[Tn + tid], pidx[Tn + tid], vA, iA);
        float vB; int iB;
        comb(pval[2 * Tn + tid], pidx[2 * Tn + tid],
             pval[3 * Tn + tid], pidx[3 * Tn + tid], vB, iB);
        float best; int bi;
        comb(vA, iA, vB, iB, best, bi);

        const bool valid = (t < len);
        const float nsc = best + pc[s * Tn + tid];
        const float old = sbuf[cur * Tn + tid];
        sbuf[(cur ^ 1) * Tn + tid] = valid ? nsc : old;
        bp[(size_t)(t - 1) * Tn + tid] = (unsigned char)(valid ? bi : tid);
      }
      __syncthreads();
      cur ^= 1;
    }
  }

  // -------- Parallel backtrace over LDS-resident backpointers --------
  // Phase A: compose per-segment tag maps (4 groups x 48 entry-tags).
  for (int g = q; g < NSEG; g += 4) {
    const int start = g * SEGLEN;
    int end = start + SEGLEN;
    if (end > NSTEPS) end = NSTEPS;
    int x = j;
    for (int s = end - 1; s >= start; --s) x = (int)bp[s * Tn + x];
    segmap[g * Tn + j] = (unsigned char)x;
  }
  __syncthreads();

  int* ot = out_tags + (size_t)b * Ln;

  // Phase B: final argmax + walk segment boundaries (thread 0).
  if (tid == 0) {
    const float* fs = sbuf + cur * Tn;
    float bv = fs[0];
    int tag = 0;
    for (int i = 1; i < Tn; ++i)
      if (fs[i] > bv) { bv = fs[i]; tag = i; }
    int x = tag;                            // tag at time L-1
    for (int g = NSEG - 1; g >= 0; --g) {
      ein[g] = x;                           // tag at time end(g)
      x = (int)segmap[g * Tn + x];          // -> tag at time start(g)
    }
    ot[0] = x;                              // tag at time 0
  }
  __syncthreads();

  // Phase C: each of 64 threads replays its segment, emitting tags.
  if (tid < NSEG) {
    const int g = tid;
    const int start = g * SEGLEN;
    int end = start + SEGLEN;
    if (end > NSTEPS) end = NSTEPS;
    int x = ein[g];
    ot[end] = x;
    for (int s = end - 1; s > start; --s) {
      x = (int)bp[s * Tn + x];
      ot[s] = x;
    }
  }
}

extern "C" void kernel_launch(void* const* d_in, const int* in_sizes, int n_in,
                              void* d_out, int out_size, void* d_ws,
                              size_t ws_size, hipStream_t stream) {
  const float* pot     = (const float*)d_in[0];   // (B, L, T) f32
  const int*   lengths = (const int*)d_in[1];     // (B,) i32
  const float* trans   = (const float*)d_in[2];   // (T, T) f32
  int*   out_tags  = (int*)d_out;                       // (B, L) i32
  float* out_trans = (float*)d_out + (size_t)Bn * Ln;   // (T, T) f32

  crf_viterbi_kernel<<<dim3(Bn), dim3(NTHREADS), LDS_BYTES, stream>>>(
      pot, lengths, trans, out_tags, out_trans);
}